// DiffGatedTopK_2181843386942
// MI455X (gfx1250) — compile-verified
//
#include <hip/hip_runtime.h>
#include <stdint.h>

#define D        4096
#define THREADS  256
#define VPT      (D / 4 / THREADS)   // float4 chunks per thread = 4
#define AS3      __attribute__((address_space(3)))

typedef float v4f __attribute__((ext_vector_type(4)));

// Order-preserving float -> u32 key (larger float <=> larger key)
__device__ __forceinline__ uint32_t f2key(float f) {
    uint32_t b = __float_as_uint(f);
    return b ^ ((uint32_t)((int32_t)b >> 31) | 0x80000000u);
}
__device__ __forceinline__ float key2f(uint32_t k) {
    uint32_t b = k ^ ((uint32_t)((int32_t)(~k) >> 31) | 0x80000000u);
    return __uint_as_float(b);
}

__global__ __launch_bounds__(THREADS)
void DiffGatedTopK_kernel(const float* __restrict__ x,
                          float* __restrict__ out,
                          int k) {
    __shared__ __align__(16) float srow[D];
    __shared__ uint32_t hist[256];
    __shared__ uint32_t s_w1[THREADS / 32], s_w2[THREADS / 32];
    __shared__ uint32_t s_selbin, s_kk;
    __shared__ float    s_gain;

    const int row = blockIdx.x;
    const int tid = threadIdx.x;
    const float* gx = x   + (size_t)row * D;
    float*       gy = out + (size_t)row * D;

    // ---- Stage row into LDS via CDNA5 async copy engine (ASYNCcnt path) ----
    uint32_t lds_base = (uint32_t)(size_t)(AS3 float*)srow;
#pragma unroll
    for (int r = 0; r < VPT; ++r) {
        int v = r * THREADS + tid;                 // float4 index, coalesced per round
        uint64_t ga  = (uint64_t)(gx + 4 * v);
        uint32_t lds = lds_base + 16u * (uint32_t)v;
        asm volatile("global_load_async_to_lds_b128 %0, %1, off"
                     :: "v"(lds), "v"(ga) : "memory");
    }
#if __has_builtin(__builtin_amdgcn_s_wait_asynccnt)
    __builtin_amdgcn_s_wait_asynccnt(0);
#else
    asm volatile("s_wait_asynccnt 0" ::: "memory");
#endif
    __syncthreads();

    // ---- Top-2 reduction (keys), wave32 shuffles + cross-wave LDS merge ----
    uint32_t m1 = 0u, m2 = 0u;
#pragma unroll
    for (int r = 0; r < VPT; ++r) {
        int v = r * THREADS + tid;
        v4f f = ((const v4f*)srow)[v];
#pragma unroll
        for (int c = 0; c < 4; ++c) {
            uint32_t key = f2key(f[c]);
            uint32_t lo  = key < m1 ? key : m1;    // min(m1, key)
            m1 = key > m1 ? key : m1;
            m2 = lo > m2 ? lo : m2;
        }
    }
#pragma unroll
    for (int off = 16; off >= 1; off >>= 1) {      // full wave32 butterfly
        uint32_t o1 = __shfl_xor(m1, off, 32);
        uint32_t o2 = __shfl_xor(m2, off, 32);
        uint32_t lo = m1 < o1 ? m1 : o1;
        uint32_t hi2 = m2 > o2 ? m2 : o2;
        m1 = m1 > o1 ? m1 : o1;
        m2 = lo > hi2 ? lo : hi2;
    }
    if ((tid & 31) == 0) { s_w1[tid >> 5] = m1; s_w2[tid >> 5] = m2; }
    __syncthreads();
    if (tid == 0) {
        uint32_t a1 = s_w1[0], a2 = s_w2[0];
        for (int w = 1; w < THREADS / 32; ++w) {
            uint32_t o1 = s_w1[w], o2 = s_w2[w];
            uint32_t lo = a1 < o1 ? a1 : o1;
            uint32_t hi2 = a2 > o2 ? a2 : o2;
            a1 = a1 > o1 ? a1 : o1;
            a2 = lo > hi2 ? lo : hi2;
        }
        float diff = key2f(a1) - key2f(a2);
        float sig  = 1.0f / (1.0f + __expf(-diff));
        s_gain = sig * 3.0f + 1.0f;
    }

    // ---- 4-pass radix select: exact k-th largest key of the row ----
    uint32_t prefix_key = 0u, prefix_mask = 0u;
    uint32_t kk = (uint32_t)k;
#pragma unroll 1
    for (int byte = 3; byte >= 0; --byte) {
        const int sh = byte * 8;
        hist[tid] = 0u;
        __syncthreads();
#pragma unroll
        for (int r = 0; r < VPT; ++r) {
            int v = r * THREADS + tid;
            v4f f = ((const v4f*)srow)[v];
#pragma unroll
            for (int c = 0; c < 4; ++c) {
                uint32_t key = f2key(f[c]);
                if ((key & prefix_mask) == prefix_key)
                    atomicAdd(&hist[(key >> sh) & 0xFFu], 1u);
            }
        }
        __syncthreads();
        // inclusive suffix scan over 256 bins (Hillis-Steele)
#pragma unroll 1
        for (int off = 1; off < 256; off <<= 1) {
            uint32_t val = hist[tid];
            uint32_t add = (tid + off < 256) ? hist[tid + off] : 0u;
            __syncthreads();
            hist[tid] = val + add;
            __syncthreads();
        }
        uint32_t sfx   = hist[tid];
        uint32_t above = (tid == 255) ? 0u : hist[tid + 1];
        if (sfx >= kk && above < kk) {             // exactly one thread matches
            s_selbin = (uint32_t)tid;
            s_kk     = kk - above;
        }
        __syncthreads();
        uint32_t b = s_selbin;
        kk = s_kk;
        prefix_key  |= b << sh;
        prefix_mask |= 0xFFu << sh;
        __syncthreads();
    }
    const uint32_t thresh = prefix_key;            // key of k-th largest element
    const float gain = s_gain;

    // ---- Output pass: gate + scale, nontemporal B128 stores ----
#pragma unroll
    for (int r = 0; r < VPT; ++r) {
        int v = r * THREADS + tid;
        v4f f = ((const v4f*)srow)[v];
        v4f o;
#pragma unroll
        for (int c = 0; c < 4; ++c) {
            uint32_t key = f2key(f[c]);
            o[c] = (key >= thresh) ? f[c] * gain : 0.0f;
        }
        __builtin_nontemporal_store(o, (v4f*)(gy + 4 * v));
    }
}

extern "C" void kernel_launch(void* const* d_in, const int* in_sizes, int n_in,
                              void* d_out, int out_size, void* d_ws, size_t ws_size,
                              hipStream_t stream) {
    const float* x = (const float*)d_in[0];
    float* out = (float*)d_out;
    const int rows = in_sizes[0] / D;              // 16384
    int k = (int)((float)D * 0.15f);               // 614
    if (k < 2) k = 2;
    DiffGatedTopK_kernel<<<dim3(rows), dim3(THREADS), 0, stream>>>(x, out, k);
}